// LossFrameAlignedGraph_40819369181408
// MI455X (gfx1250) — compile-verified
//
#include <hip/hip_runtime.h>
#include <math.h>

#define NRES 2048
#define BATCH 2
#define KNN 30
#define KEDGE 60
#define FEPS 0.01f

typedef float v2f __attribute__((ext_vector_type(2)));
typedef float v8f __attribute__((ext_vector_type(8)));

// ---------------------------------------------------------------------------
// Kernel 1: pairwise squared distances of CA atoms via V_WMMA_F32_16X16X4_F32.
// One wave computes a 16x128 strip: the A fragment (16 rows of points) and the
// 8 per-row norms/masks are loaded ONCE, then 8 unrolled WMMAs sweep 8 column
// tiles. d2 = |pi|^2 + |pj|^2 - 2*pi.pj, Gram dot by f32 WMMA with K=4
// (4th component zero-padded).
//   A layout (16x4 MxK): VGPR v, lane L -> K = v + 2*(L>=16), M = L&15
//   B layout (4x16 KxN): VGPR v, lane L -> K = v + 2*(L>=16), N = L&15
//   D layout (16x16):    VGPR v, lane L -> M = v + 8*(L>=16), N = L&15
// No EXEC divergence anywhere (WMMA requires EXEC == all ones).
// ---------------------------------------------------------------------------
__global__ __launch_bounds__(32) void d2_wmma_kernel(
    const float* __restrict__ Xs,   // (NRES, 4, 3) for one (batch, source)
    const int*   __restrict__ Cb,   // (NRES) residue mask for this batch
    float*       __restrict__ d2)   // (NRES, NRES) output
{
    __shared__ float normI[16];

    const int lane = threadIdx.x;        // 0..31
    const int half = lane >> 4;          // 0 = lanes 0-15, 1 = lanes 16-31
    const int l16  = lane & 15;
    const int m0   = blockIdx.y << 4;    // 16-row strip
    const int nb   = blockIdx.x << 7;    // 128-column group (8 tiles)

    // ---- A side, loaded once per wave ------------------------------------
    // CA atom = atom index 1 (each residue is 12 floats)
    const float* Pi = Xs + (size_t)(m0 + l16) * 12 + 3;
    const float pix = Pi[0], piy = Pi[1], piz = Pi[2];

    // Both halves write the same value (same l16 -> same point); benign.
    normI[l16] = pix * pix + piy * piy + piz * piz;
    __syncthreads();

    v2f a;                               // K = v + 2*half ; pad K=3 with 0
    a.x = half ? piz : pix;
    a.y = half ? 0.0f : piy;

    float ni2[8];
    bool  ciok[8];
#pragma unroll
    for (int v = 0; v < 8; ++v) {
        const int mrow = v + (half << 3);
        ni2[v]  = normI[mrow];
        ciok[v] = (Cb[m0 + mrow] > 0);
    }

    // ---- 8 column tiles, one WMMA each -----------------------------------
#pragma unroll
    for (int t = 0; t < 8; ++t) {
        const int n0 = nb + (t << 4);
        const float* Pj = Xs + (size_t)(n0 + l16) * 12 + 3;
        const float pjx = Pj[0], pjy = Pj[1], pjz = Pj[2];
        const float nj2 = pjx * pjx + pjy * pjy + pjz * pjz;

        v2f b;
        b.x = half ? pjz : pjx;
        b.y = half ? 0.0f : pjy;

        v8f c = {};
        c = __builtin_amdgcn_wmma_f32_16x16x4_f32(
                /*neg_a=*/false, a, /*neg_b=*/false, b,
                /*c_mod=*/(short)0, c, /*reuse_a=*/false, /*reuse_b=*/false);

        const bool cjok = (Cb[n0 + l16] > 0);

#pragma unroll
        for (int v = 0; v < 8; ++v) {
            const int mrow = v + (half << 3);
            float d = ni2[v] + nj2 - 2.0f * c[v];
            d = (ciok[v] && cjok) ? d : 1000000000.0f;   // masked pairs -> 1e9
            d2[(size_t)(m0 + mrow) * NRES + (n0 + l16)] = d;
        }
    }
}

// ---------------------------------------------------------------------------
// Kernel 2: stable top-K (K=30) smallest per row (ties -> lower index first),
// matching jax.lax.top_k(-d2, 30) semantics. One thread per row.
// ---------------------------------------------------------------------------
__global__ void topk_kernel(const float* __restrict__ d2,
                            int* __restrict__ edgesB,   // (NRES, 60), batch slice
                            int srcOff)                 // 0 for X, 30 for X_target
{
    const int i = blockIdx.x * blockDim.x + threadIdx.x;
    if (i >= NRES) return;

    float bd[KNN];
    int   bi[KNN];
#pragma unroll
    for (int k = 0; k < KNN; ++k) { bd[k] = 3.0e38f; bi[k] = 0; }

    const float* row = d2 + (size_t)i * NRES;
    for (int j = 0; j < NRES; ++j) {
        const float d = row[j];
        if (d < bd[KNN - 1]) {                // strict: equal keeps earlier index
            int p = KNN - 1;
            while (p > 0 && bd[p - 1] > d) {  // strict: insert after equals
                bd[p] = bd[p - 1];
                bi[p] = bi[p - 1];
                --p;
            }
            bd[p] = d;
            bi[p] = j;
        }
    }

    int* e = edgesB + (size_t)i * KEDGE + srcOff;
#pragma unroll
    for (int k = 0; k < KNN; ++k) e[k] = bi[k];
}

// ---------------------------------------------------------------------------
// Kernel 3: frame-aligned distance loss.
// ---------------------------------------------------------------------------
__device__ __forceinline__ void make_frame(const float* __restrict__ bb, // 12 floats: N,CA,C,(O)
                                           float n1[3], float n2[3], float n3[3],
                                           float ca[3])
{
    const float Nx = bb[0], Ny = bb[1], Nz = bb[2];
    const float Ax = bb[3], Ay = bb[4], Az = bb[5];
    const float Cx = bb[6], Cy = bb[7], Cz = bb[8];

    float ux = Nx - Ax, uy = Ny - Ay, uz = Nz - Az;
    float inv = 1.0f / sqrtf(ux * ux + uy * uy + uz * uz + FEPS);
    n1[0] = ux * inv; n1[1] = uy * inv; n1[2] = uz * inv;

    float vx = Cx - Ax, vy = Cy - Ay, vz = Cz - Az;
    inv = 1.0f / sqrtf(vx * vx + vy * vy + vz * vz + FEPS);
    vx *= inv; vy *= inv; vz *= inv;

    const float p = vx * n1[0] + vy * n1[1] + vz * n1[2];
    float tx = vx - p * n1[0], ty = vy - p * n1[1], tz = vz - p * n1[2];
    inv = 1.0f / sqrtf(tx * tx + ty * ty + tz * tz + FEPS);
    n2[0] = tx * inv; n2[1] = ty * inv; n2[2] = tz * inv;

    n3[0] = n1[1] * n2[2] - n1[2] * n2[1];
    n3[1] = n1[2] * n2[0] - n1[0] * n2[2];
    n3[2] = n1[0] * n2[1] - n1[1] * n2[0];

    ca[0] = Ax; ca[1] = Ay; ca[2] = Az;
}

__global__ void loss_kernel(const float* __restrict__ X,
                            const float* __restrict__ Xt,
                            const int*   __restrict__ C,
                            const int*   __restrict__ edges,  // (BATCH, NRES, 60)
                            float*       __restrict__ out)    // (BATCH, NRES)
{
    const int t = blockIdx.x * blockDim.x + threadIdx.x;
    if (t >= BATCH * NRES) return;
    const int b = t / NRES;

    const float mi = (C[t] > 0) ? 1.0f : 0.0f;
    const float* Xi  = X  + (size_t)t * 12;
    const float* Xti = Xt + (size_t)t * 12;
    const int*   e   = edges + (size_t)t * KEDGE;

    float Dsum = 0.0f, Msum = 0.0f;

    for (int k = 0; k < KEDGE; ++k) {
        const int j = e[k];
        const float mj  = (C[b * NRES + j] > 0) ? 1.0f : 0.0f;
        const float mij = mi * mj;

        const float* bbj  = X  + ((size_t)b * NRES + j) * 12;
        const float* bbtj = Xt + ((size_t)b * NRES + j) * 12;

        float n1[3], n2[3], n3[3], ca[3];
        float m1[3], m2[3], m3[3], cb[3];
        make_frame(bbj,  n1, n2, n3, ca);
        make_frame(bbtj, m1, m2, m3, cb);

#pragma unroll
        for (int a = 0; a < 4; ++a) {
            const float dx = Xi[a * 3 + 0] - ca[0];
            const float dy = Xi[a * 3 + 1] - ca[1];
            const float dz = Xi[a * 3 + 2] - ca[2];
            const float r0 = dx * n1[0] + dy * n1[1] + dz * n1[2];
            const float r1 = dx * n2[0] + dy * n2[1] + dz * n2[2];
            const float r2 = dx * n3[0] + dy * n3[1] + dz * n3[2];

            const float ex = Xti[a * 3 + 0] - cb[0];
            const float ey = Xti[a * 3 + 1] - cb[1];
            const float ez = Xti[a * 3 + 2] - cb[2];
            const float s0 = ex * m1[0] + ey * m1[1] + ez * m1[2];
            const float s1 = ex * m2[0] + ey * m2[1] + ez * m2[2];
            const float s2 = ex * m3[0] + ey * m3[1] + ez * m3[2];

            const float q0 = r0 - s0, q1 = r1 - s1, q2 = r2 - s2;
            Dsum += mij * sqrtf(q0 * q0 + q1 * q1 + q2 * q2 + FEPS);
            Msum += mij;
        }
    }

    out[t] = Dsum / (Msum + FEPS);
}

// ---------------------------------------------------------------------------
// Host launcher. Workspace layout:
//   [0, NRES*NRES*4)                      : d2 matrix (reused per batch/source)
//   [NRES*NRES*4, + BATCH*NRES*KEDGE*4)   : edge indices
// Kernels on the same stream serialize, so the d2 buffer is safely reused.
// ---------------------------------------------------------------------------
extern "C" void kernel_launch(void* const* d_in, const int* in_sizes, int n_in,
                              void* d_out, int out_size, void* d_ws, size_t ws_size,
                              hipStream_t stream) {
    const float* X  = (const float*)d_in[0];   // (B, N, 4, 3) f32
    const float* Xt = (const float*)d_in[1];   // (B, N, 4, 3) f32
    const int*   C  = (const int*)d_in[2];     // (B, N) i32
    float* out = (float*)d_out;                // (B, N) f32

    float* d2    = (float*)d_ws;
    int*   edges = (int*)((char*)d_ws + (size_t)NRES * NRES * sizeof(float));

    for (int b = 0; b < BATCH; ++b) {
        for (int src = 0; src < 2; ++src) {
            const float* Xs = (src ? Xt : X) + (size_t)b * NRES * 12;
            const int*   Cb = C + (size_t)b * NRES;
            // 16 column-groups x 128 row-strips, one wave per block
            d2_wmma_kernel<<<dim3(NRES / 128, NRES / 16), 32, 0, stream>>>(Xs, Cb, d2);
            topk_kernel<<<NRES / 128, 128, 0, stream>>>(
                d2, edges + (size_t)b * NRES * KEDGE, src * KNN);
        }
    }

    loss_kernel<<<(BATCH * NRES + 127) / 128, 128, 0, stream>>>(X, Xt, C, edges, out);
}